// Net_33517924778672
// MI455X (gfx1250) — compile-verified
//
#include <hip/hip_runtime.h>
#include <hip/hip_bf16.h>
#include <math.h>

typedef __attribute__((ext_vector_type(2))) float v2f;
typedef __attribute__((ext_vector_type(8))) float v8f;

#define N_NODES 100000
#define N_EDGES 1250000
#define DIM     64

// ---------------------------------------------------------------------------
// Utility: fill a float buffer with a constant
// ---------------------------------------------------------------------------
__global__ void fill_f32(float* __restrict__ p, float v, int n) {
  int i = blockIdx.x * blockDim.x + threadIdx.x;
  if (i < n) p[i] = v;
}

// ---------------------------------------------------------------------------
// Degree: deg[dst] += 1  (deg pre-filled with 1.0 for the self-loop)
// ---------------------------------------------------------------------------
__global__ void degree_accum(const int* __restrict__ dst, float* __restrict__ deg, int e) {
  int i = blockIdx.x * blockDim.x + threadIdx.x;
  if (i < e) atomicAdd(&deg[dst[i]], 1.0f);
}

__global__ void rsqrt_inplace(float* __restrict__ p, int n) {
  int i = blockIdx.x * blockDim.x + threadIdx.x;
  if (i < n) p[i] = rsqrtf(p[i]);
}

// ---------------------------------------------------------------------------
// Dense GEMM: C[N,64] = A[N,64] @ W[64,64] via V_WMMA_F32_16X16X4_F32.
// One wave -> one 16-row tile, full 64 output columns (4 accumulators,
// 16 K-steps x 4 tiles = 64 WMMAs per wave).
//
// W is staged in LDS PRE-SWIZZLED into K-pair fragments:
//   sW[(p*64 + n)*2 + j] = W[2p + j][n],  p = k/2 in [0,32), n in [0,64), j in {0,1}
// so each lane's B fragment {W[k0][n], W[k0+1][n]} is one contiguous,
// 8-byte-aligned ds_load_b64 landing directly in an even VGPR pair
// (WMMA sources must be consecutive even-aligned VGPRs -> no repack movs).
//
// f32 WMMA layouts (wave32):
//   A 16x4:  lane l<16 : M=l, {K=k0,k0+1}; lane l>=16 : M=l-16, {K=k0+2,k0+3}
//   B 4x16:  lane l<16 : N=l, {K=k0,k0+1}; lane l>=16 : N=l-16, {K=k0+2,k0+3}
//   C/D 16x16: VGPR r : lanes 0-15 -> M=r, lanes 16-31 -> M=r+8, N=lane%16
// ---------------------------------------------------------------------------
__global__ __launch_bounds__(256) void gemm_n64_wmma(const float* __restrict__ A,
                                                     const float* __restrict__ W,
                                                     float* __restrict__ C,
                                                     int nTiles) {
  __shared__ float sW[32 * DIM * 2];  // 4096 floats = 16 KB, pair-swizzled
  for (int f = threadIdx.x; f < 32 * DIM * 2; f += 256) {
    const int j = f & 1;          // element within K-pair
    const int n = (f >> 1) & 63;  // output column
    const int p = f >> 7;         // K-pair index
    sW[f] = W[(2 * p + j) * DIM + n];
  }
  __syncthreads();

  const int wave = threadIdx.x >> 5;
  const int lane = threadIdx.x & 31;
  const int tile = blockIdx.x * 8 + wave;
  if (tile >= nTiles) return;  // uniform per wave -> EXEC stays all-ones

  const int row0 = tile << 4;   // 16 rows per tile
  const int mr   = lane & 15;   // M (for A) / N (for B/C)
  const int kh   = lane >> 4;   // K-half selector

  const float* arow = A + (size_t)(row0 + mr) * DIM;

  v8f acc0 = {}, acc1 = {}, acc2 = {}, acc3 = {};

#pragma unroll
  for (int kk = 0; kk < 16; ++kk) {
    const int k0 = kk * 4 + kh * 2;
    v2f a;
    a.x = arow[k0];
    a.y = arow[k0 + 1];

    const int p = (k0 >> 1);                   // K-pair index for this lane
    const float* wp = sW + ((p * DIM) << 1);   // base of this pair's row
    const v2f b0 = *(const v2f*)(wp + ((0  + mr) << 1));
    const v2f b1 = *(const v2f*)(wp + ((16 + mr) << 1));
    const v2f b2 = *(const v2f*)(wp + ((32 + mr) << 1));
    const v2f b3 = *(const v2f*)(wp + ((48 + mr) << 1));

    acc0 = __builtin_amdgcn_wmma_f32_16x16x4_f32(false, a, false, b0, (short)0, acc0, false, false);
    acc1 = __builtin_amdgcn_wmma_f32_16x16x4_f32(false, a, false, b1, (short)0, acc1, false, false);
    acc2 = __builtin_amdgcn_wmma_f32_16x16x4_f32(false, a, false, b2, (short)0, acc2, false, false);
    acc3 = __builtin_amdgcn_wmma_f32_16x16x4_f32(false, a, false, b3, (short)0, acc3, false, false);
  }

#pragma unroll
  for (int r = 0; r < 8; ++r) {
    const size_t m = (size_t)(row0 + r + kh * 8) * DIM;
    C[m + 0  + mr] = acc0[r];
    C[m + 16 + mr] = acc1[r];
    C[m + 32 + mr] = acc2[r];
    C[m + 48 + mr] = acc3[r];
  }
}

// ---------------------------------------------------------------------------
// Edge scatter: agg[dst, :] += xw[src, :] * (inv[src]*inv[dst])
// 16 threads per edge, float4 gathers, f32 atomics (resolved in L2).
// ---------------------------------------------------------------------------
__global__ void scatter_edges(const int* __restrict__ src, const int* __restrict__ dst,
                              const float* __restrict__ inv, const float* __restrict__ xw,
                              float* __restrict__ agg, int e) {
  const long tid = (long)blockIdx.x * blockDim.x + threadIdx.x;
  const int ei = (int)(tid >> 4);
  const int q  = (int)(tid & 15);
  if (ei >= e) return;
  const int s = src[ei], d = dst[ei];
  const float coef = inv[s] * inv[d];
  const float4 v = *(const float4*)(xw + (size_t)s * DIM + q * 4);
  float* base = agg + (size_t)d * DIM + q * 4;
  atomicAdd(base + 0, v.x * coef);
  atomicAdd(base + 1, v.y * coef);
  atomicAdd(base + 2, v.z * coef);
  atomicAdd(base + 3, v.w * coef);
}

// ---------------------------------------------------------------------------
// Layer-1 finalize: h = relu(agg + xw*inv^2 + b)   (in place into agg)
// ---------------------------------------------------------------------------
__global__ void finalize_relu(float* __restrict__ agg, const float* __restrict__ xw,
                              const float* __restrict__ inv, const float* __restrict__ bias,
                              int n) {
  const int i = blockIdx.x * blockDim.x + threadIdx.x;
  if (i >= n) return;
  const int node = i >> 6;
  const int d    = i & 63;
  const float id = inv[node];
  const float v  = agg[i] + xw[i] * (id * id) + bias[d];
  agg[i] = v > 0.0f ? v : 0.0f;
}

// ---------------------------------------------------------------------------
// Layer-2 finalize + log_softmax over the 64 features of each node.
// One thread per node (64 floats in registers).
// ---------------------------------------------------------------------------
__global__ void finalize_lsm(const float* __restrict__ agg, const float* __restrict__ xw,
                             const float* __restrict__ inv, const float* __restrict__ bias,
                             float* __restrict__ out, int n) {
  const int node = blockIdx.x * blockDim.x + threadIdx.x;
  if (node >= n) return;
  const float id = inv[node];
  const float sl = id * id;
  const size_t base = (size_t)node * DIM;
  float z[DIM];
  float m = -INFINITY;
#pragma unroll
  for (int d = 0; d < DIM; ++d) {
    z[d] = agg[base + d] + xw[base + d] * sl + bias[d];
    m = fmaxf(m, z[d]);
  }
  float s = 0.0f;
#pragma unroll
  for (int d = 0; d < DIM; ++d) s += __expf(z[d] - m);
  const float lse = m + __logf(s);
#pragma unroll
  for (int d = 0; d < DIM; ++d) out[base + d] = z[d] - lse;
}

// ---------------------------------------------------------------------------
// Host-side orchestration
// ---------------------------------------------------------------------------
extern "C" void kernel_launch(void* const* d_in, const int* in_sizes, int n_in,
                              void* d_out, int out_size, void* d_ws, size_t ws_size,
                              hipStream_t stream) {
  const float* x   = (const float*)d_in[0];
  const int*   ei  = (const int*)d_in[1];
  const float* W1  = (const float*)d_in[2];
  const float* b1  = (const float*)d_in[3];
  const float* W2  = (const float*)d_in[4];
  const float* b2  = (const float*)d_in[5];
  float*       out = (float*)d_out;

  const int* src = ei;             // edge_index[0, :]
  const int* dst = ei + N_EDGES;   // edge_index[1, :]

  // Workspace layout (all f32):
  //   inv : N floats            (degree -> inv_sqrt, 512KB slot)
  //   B0  : N*64 (xw for current layer)
  //   B1  : N*64 (agg1 -> h)
  //   B2  : N*64 (agg2)
  char* ws = (char*)d_ws;
  const size_t FEAT_BYTES = (size_t)N_NODES * DIM * sizeof(float); // 25.6 MB
  float* inv = (float*)ws;
  float* B0  = (float*)(ws + (1 << 19));
  float* B1  = (float*)(ws + (1 << 19) + FEAT_BYTES);
  float* B2  = (float*)(ws + (1 << 19) + 2 * FEAT_BYTES);

  const int NF       = N_NODES * DIM;                  // 6.4M
  const int nTiles   = N_NODES / 16;                   // 6250 (exact)
  const int gemmGrid = (nTiles + 7) / 8;               // 8 waves per block
  const long scatT   = (long)N_EDGES * 16;             // threads for scatter
  const int scatGrid = (int)((scatT + 255) / 256);

  // 1) deg = 1.0 (self-loop), accumulate in-degree, then inv_sqrt
  fill_f32<<<(N_NODES + 255) / 256, 256, 0, stream>>>(inv, 1.0f, N_NODES);
  degree_accum<<<(N_EDGES + 255) / 256, 256, 0, stream>>>(dst, inv, N_EDGES);
  rsqrt_inplace<<<(N_NODES + 255) / 256, 256, 0, stream>>>(inv, N_NODES);

  // 2) Layer 1: xw1 = x @ W1 ; agg1 = scatter ; h = relu(agg1 + xw1*inv^2 + b1)
  gemm_n64_wmma<<<gemmGrid, 256, 0, stream>>>(x, W1, B0, nTiles);
  fill_f32<<<(NF + 255) / 256, 256, 0, stream>>>(B1, 0.0f, NF);
  scatter_edges<<<scatGrid, 256, 0, stream>>>(src, dst, inv, B0, B1, N_EDGES);
  finalize_relu<<<(NF + 255) / 256, 256, 0, stream>>>(B1, B0, inv, b1, NF);

  // 3) Layer 2: xw2 = h @ W2 ; agg2 = scatter ; out = log_softmax(...)
  gemm_n64_wmma<<<gemmGrid, 256, 0, stream>>>(B1, W2, B0, nTiles);
  fill_f32<<<(NF + 255) / 256, 256, 0, stream>>>(B2, 0.0f, NF);
  scatter_edges<<<scatGrid, 256, 0, stream>>>(src, dst, inv, B0, B2, N_EDGES);
  finalize_lsm<<<(N_NODES + 255) / 256, 256, 0, stream>>>(B2, B0, inv, b2, out, N_NODES);
}